// ClusteringLayer_57131654972013
// MI455X (gfx1250) — compile-verified
//
#include <hip/hip_runtime.h>

#define N_CLUSTERS 256
#define EMBED_DIM  128
#define LDS_STRIDE 132          // 128 + 4 pad -> conflict-free ds_load_b64 fragments
#define ROWS_PER_BLOCK 128
#define TILES_PER_BLOCK 4

typedef __attribute__((ext_vector_type(2))) float v2f;
typedef __attribute__((ext_vector_type(8))) float v8f;

// LDS layout (byte offsets from dynamic-LDS base, which is LDS offset 0 here)
#define LDS_C_FLOATS   (N_CLUSTERS * LDS_STRIDE)
#define LDS_X_FLOATS   (ROWS_PER_BLOCK * LDS_STRIDE)
#define LDS_X0_OFF     (LDS_C_FLOATS * 4)
#define LDS_X1_OFF     (LDS_X0_OFF + LDS_X_FLOATS * 4)
#define LDS_CSQ_OFF    (LDS_X1_OFF + LDS_X_FLOATS * 4)
#define LDS_BYTES      (LDS_CSQ_OFF + N_CLUSTERS * 4)

// Issue one X row-tile (128 rows x 128 f32) as async global->LDS b128 copies.
// 4096 16B chunks, 16 per thread; tracked on ASYNCcnt, no data VGPRs consumed.
__device__ __forceinline__ void async_stage_x(const float* __restrict__ X,
                                              int nrows, int row0,
                                              unsigned lds_byte_off, int tid)
{
    #pragma unroll
    for (int it = 0; it < 16; ++it) {
        int i = tid + it * 256;
        int r = i >> 5, c = i & 31;
        int gr = row0 + r; if (gr >= nrows) gr = nrows - 1;
        unsigned long long gp =
            (unsigned long long)(size_t)(X + (size_t)gr * EMBED_DIM + c * 4);
        unsigned lo = lds_byte_off + (unsigned)(r * (LDS_STRIDE * 4) + c * 16);
        asm volatile("global_load_async_to_lds_b128 %0, %1, off"
                     :: "v"(lo), "v"(gp) : "memory");
    }
}

__device__ __forceinline__ void wait_async_all() {
    asm volatile("s_wait_asynccnt 0" ::: "memory");
}

__global__ __launch_bounds__(256) void dec_cluster_kernel(
    const float* __restrict__ X, const float* __restrict__ C,
    float* __restrict__ Q, int nrows, int ntiles)
{
    extern __shared__ float lds[];
    float* ldsC    = lds;                                    // 256 * 132
    float* ldsX0   = lds + LDS_C_FLOATS;                     // 128 * 132
    float* ldsX1   = ldsX0 + LDS_X_FLOATS;                   // 128 * 132
    float* ldsCsq  = ldsX1 + LDS_X_FLOATS;                   // 256

    const int tid  = threadIdx.x;
    const int lane = tid & 31;
    const int wave = tid >> 5;          // 0..7
    const int half = lane >> 4;         // 0/1
    const int l16  = lane & 15;
    const int rbase = wave * 16;

    // ---- Kick off async load of this block's first X tile immediately ----
    int t0 = blockIdx.x;
    if (t0 < ntiles)
        async_stage_x(X, nrows, t0 * ROWS_PER_BLOCK, LDS_X0_OFF, tid);

    // ---- Stage clusters into LDS once per block (coalesced, padded rows) ----
    for (int i = tid; i < N_CLUSTERS * (EMBED_DIM / 4); i += 256) {
        int r = i >> 5, c = i & 31;
        ((float4*)(ldsC + r * LDS_STRIDE))[c] = ((const float4*)C)[i];
    }
    __syncthreads();

    // ---- ||c||^2 per cluster (one cluster per thread; <=2-way conflicts) ----
    {
        const float4* cp = (const float4*)(ldsC + tid * LDS_STRIDE);
        float s = 0.f;
        #pragma unroll
        for (int k = 0; k < EMBED_DIM / 4; ++k) {
            float4 v = cp[k];
            s += v.x * v.x + v.y * v.y + v.z * v.z + v.w * v.w;
        }
        ldsCsq[tid] = s;
    }
    wait_async_all();       // first X tile landed
    __syncthreads();

    // ---- Persistent tile loop with double-buffered async X prefetch ----
    int cur = 0;
    for (int t = t0; t < ntiles; t += gridDim.x) {
        int tn = t + gridDim.x;
        if (tn < ntiles)     // prefetch next tile into the other buffer
            async_stage_x(X, nrows, tn * ROWS_PER_BLOCK,
                          cur ? LDS_X0_OFF : LDS_X1_OFF, tid);

        const int row0 = t * ROWS_PER_BLOCK;
        const float* ldsX = cur ? ldsX1 : ldsX0;   // register select, no scratch

        // ---- WMMA core: wave owns 16 rows x 256 clusters ----
        v8f acc[16];
        #pragma unroll
        for (int j = 0; j < 16; ++j)
            #pragma unroll
            for (int e = 0; e < 8; ++e) acc[j][e] = 0.f;

        // A frag: lane(h, m=l16) holds K = k0+2h, k0+2h+1 of row rbase+l16
        const float* aptr = ldsX + (rbase + l16) * LDS_STRIDE + half * 2;
        // B frag: lane(h, n=l16) holds clusters[j*16+l16][k0+2h .. +1]
        const float* bptr = ldsC + l16 * LDS_STRIDE + half * 2;

        float xsq = 0.f;
        #pragma unroll 2
        for (int k0 = 0; k0 < EMBED_DIM; k0 += 4) {
            v2f a = *(const v2f*)(aptr + k0);
            xsq += a.x * a.x + a.y * a.y;
            #pragma unroll
            for (int j = 0; j < 16; ++j) {
                v2f b = *(const v2f*)(bptr + j * 16 * LDS_STRIDE + k0);
                acc[j] = __builtin_amdgcn_wmma_f32_16x16x4_f32(
                    false, a, false, b, (short)0, acc[j], false, false);
            }
        }

        // combine K-halves of ||x||^2: lanes l, l^16 together cover row l16
        xsq += __shfl_xor(xsq, 16, 32);
        // distribute to C/D layout rows: VGPR r -> row rbase + r + 8*half
        float xs[8];
        #pragma unroll
        for (int r = 0; r < 8; ++r) xs[r] = __shfl(xsq, r + half * 8, 32);

        // ---- q = 1/(1+d^2) via v_rcp_f32 (~1 ulp), accumulate row sums ----
        float qsum[8];
        #pragma unroll
        for (int r = 0; r < 8; ++r) qsum[r] = 0.f;

        #pragma unroll
        for (int j = 0; j < 16; ++j) {
            float csq = ldsCsq[j * 16 + l16];
            #pragma unroll
            for (int r = 0; r < 8; ++r) {
                float d2 = fmaxf(xs[r] + csq - 2.f * acc[j][r], 0.f);
                float q  = __builtin_amdgcn_rcpf(1.f + d2);
                acc[j][r] = q;
                qsum[r] += q;
            }
        }
        // row sum across the 16 lanes of each half
        #pragma unroll
        for (int r = 0; r < 8; ++r) {
            float s = qsum[r];
            s += __shfl_xor(s, 1, 32);
            s += __shfl_xor(s, 2, 32);
            s += __shfl_xor(s, 4, 32);
            s += __shfl_xor(s, 8, 32);
            qsum[r] = __builtin_amdgcn_rcpf(s);
        }

        // ---- normalize + non-temporal store (Q is bigger than L2) ----
        #pragma unroll
        for (int r = 0; r < 8; ++r) {
            int gr = row0 + rbase + r + half * 8;
            if (gr < nrows) {
                float* out = Q + (size_t)gr * N_CLUSTERS + l16;
                float invs = qsum[r];
                #pragma unroll
                for (int j = 0; j < 16; ++j)
                    __builtin_nontemporal_store(acc[j][r] * invs, out + j * 16);
            }
        }

        wait_async_all();    // prefetched tile landed (overlapped with compute)
        __syncthreads();     // all waves done reading cur + all async done
        cur ^= 1;
    }
}

extern "C" void kernel_launch(void* const* d_in, const int* in_sizes, int n_in,
                              void* d_out, int out_size, void* d_ws, size_t ws_size,
                              hipStream_t stream) {
    const float* X = (const float*)d_in[0];
    const float* C = (const float*)d_in[1];
    float* Q = (float*)d_out;

    int nrows  = in_sizes[0] / EMBED_DIM;
    int ntiles = (nrows + ROWS_PER_BLOCK - 1) / ROWS_PER_BLOCK;
    int grid   = (ntiles + TILES_PER_BLOCK - 1) / TILES_PER_BLOCK;
    if (grid < 1) grid = 1;

    hipFuncSetAttribute((const void*)dec_cluster_kernel,
                        hipFuncAttributeMaxDynamicSharedMemorySize,
                        (int)LDS_BYTES);

    hipLaunchKernelGGL(dec_cluster_kernel, dim3(grid), dim3(256), LDS_BYTES,
                       stream, X, C, Q, nrows, ntiles);
}